// DDPMITERVAEQueryEncoder_24172075942238
// MI455X (gfx1250) — compile-verified
//
#include <hip/hip_runtime.h>
#include <stdint.h>

#define DD 64
#define TSTEPS 200
#define BB 8192
#define LL 100
#define ROWS_PER_BLOCK 32   // 2 waves/block, 16 rows per wave

typedef float v2f __attribute__((ext_vector_type(2)));
typedef float v8f __attribute__((ext_vector_type(8)));

__device__ __forceinline__ v8f vzero8() {
  v8f v;
#pragma unroll
  for (int i = 0; i < 8; ++i) v[i] = 0.0f;
  return v;
}

// ---- deterministic counter-based PRNG (hash -> Box-Muller normal) ----
__device__ __forceinline__ uint32_t hash3(uint32_t a, uint32_t b, uint32_t c) {
  uint32_t s = a * 0x9E3779B9u + b * 0x85EBCA6Bu + c * 0xC2B2AE35u + 0x27D4EB2Fu;
  s ^= s >> 16; s *= 0x7FEB352Du;
  s ^= s >> 15; s *= 0x846CA68Bu;
  s ^= s >> 16;
  return s;
}
__device__ __forceinline__ float rng_normal(uint32_t a, uint32_t b, uint32_t c) {
  uint32_t s = hash3(a, b, c);
  uint32_t t = hash3(s, c + 0x5851F42Du, a ^ 0x14057B7Eu);
  float u1 = ((s >> 8) + 1u) * (1.0f / 16777217.0f);   // (0,1]
  float u2 = (t >> 8) * (1.0f / 16777216.0f);          // [0,1)
  return sqrtf(-2.0f * __logf(u1)) * __cosf(6.28318530718f * u2);
}

// ============ K1: gather + pooled sum / sqrt(count) ============
__global__ void gather_pool_kernel(const int* __restrict__ seq,
                                   const float* __restrict__ emb,
                                   float* __restrict__ pooled) {
  int wave = (blockIdx.x * blockDim.x + threadIdx.x) >> 5;  // one wave per row
  int lane = threadIdx.x & 31;
  if (wave >= BB) return;
  const int* s = seq + wave * LL;
  float ax = 0.0f, ay = 0.0f;
  int cnt = 0;
  for (int l = 0; l < LL; ++l) {
    int idx = s[l];
    cnt += (idx != 0) ? 1 : 0;
    const float2* e = (const float2*)(emb + (size_t)idx * DD) + lane;
    float2 v = *e;
    ax += v.x; ay += v.y;
  }
  float sc = rsqrtf((float)cnt);
  float2* op = (float2*)(pooled + (size_t)wave * DD) + lane;
  *op = make_float2(ax * sc, ay * sc);
}

// ============ K2: encoder MLP -> z (=mu) -> condproj = z@W_c + b_c ============
__global__ void encoder_kernel(const float* __restrict__ pooled,
                               const float* __restrict__ W1, const float* __restrict__ b1,
                               const float* __restrict__ W2, const float* __restrict__ b2,
                               const float* __restrict__ Wc, const float* __restrict__ bc,
                               float* __restrict__ cp) {
  __shared__ float p[DD];
  __shared__ float h1[256];
  __shared__ float zz[DD];
  int row = blockIdx.x;
  int tid = threadIdx.x;                 // 256 threads
  if (tid < DD) p[tid] = pooled[(size_t)row * DD + tid];
  __syncthreads();
  float s = b1[tid];
#pragma unroll 8
  for (int k = 0; k < DD; ++k) s += p[k] * W1[k * 256 + tid];
  h1[tid] = fmaxf(s, 0.0f);
  __syncthreads();
  if (tid < DD) {
    float e = b2[tid];                   // mu = enc[:, :64]
#pragma unroll 8
    for (int k = 0; k < 256; ++k) e += h1[k] * W2[k * 128 + tid];
    zz[tid] = e;
  }
  __syncthreads();
  if (tid < DD) {
    float c = bc[tid];
#pragma unroll 8
    for (int k = 0; k < DD; ++k) c += zz[k] * Wc[k * DD + tid];
    cp[(size_t)row * DD + tid] = c;
  }
}

// ============ K3: diffusion schedule (serial cumprod, tiny) ============
__global__ void schedule_kernel(float* __restrict__ sched) {
  if (threadIdx.x != 0 || blockIdx.x != 0) return;
  const float scale = 1000.0f / TSTEPS;
  const float b0 = scale * 1e-4f, b1 = scale * 0.02f;
  float acp = 1.0f;
  for (int t = 0; t < TSTEPS; ++t) {
    float beta = b0 + (b1 - b0) * ((float)t / (float)(TSTEPS - 1));
    float alpha = 1.0f - beta;
    float acp_prev = acp;
    acp = acp * alpha;
    sched[t]             = sqrtf(1.0f / acp);                         // sqrt_recip
    sched[200 + t]       = sqrtf(1.0f / acp - 1.0f);                  // sqrt_recipm1
    float post_var       = beta * (1.0f - acp_prev) / (1.0f - acp);
    sched[400 + t]       = beta * sqrtf(acp_prev) / (1.0f - acp);     // coef1
    sched[600 + t]       = (1.0f - acp_prev) * sqrtf(alpha) / (1.0f - acp); // coef2
    sched[800 + t]       = (t > 0) ? sqrtf(post_var) : 0.0f;          // sigma (FIXED_SMALL)
  }
}

// ============ K4: tv[t] = temb(t)@W_t + b_t + b_in ============
__global__ void temb_kernel(const float* __restrict__ Wt, const float* __restrict__ bt,
                            const float* __restrict__ bin, float* __restrict__ tv) {
  __shared__ float te[DD];
  int t = blockIdx.x;
  int d = threadIdx.x;                   // 64 threads
  {
    int i = d & 31;
    float fr = __expf(-logf(10000.0f) * (float)i / 32.0f);
    float ang = (float)t * fr;
    te[d] = (d < 32) ? __cosf(ang) : __sinf(ang);
  }
  __syncthreads();
  float s = bt[d] + bin[d];
#pragma unroll 8
  for (int k = 0; k < DD; ++k) s += te[k] * Wt[k * DD + d];
  tv[t * DD + d] = s;
}

// ============ K5: persistent p_sample_loop with f32 WMMA ============
__global__ __launch_bounds__(64) void diffusion_kernel(
    const float* __restrict__ W_in, const float* __restrict__ W_out,
    const float* __restrict__ cp, const float* __restrict__ tv,
    const float* __restrict__ sched, const float* __restrict__ b_out,
    float* __restrict__ out) {
  __shared__ float wi[DD * DD];
  __shared__ float wo[DD * DD];
  __shared__ float xs[ROWS_PER_BLOCK * DD];
  __shared__ float hs[ROWS_PER_BLOCK * DD];

  int tid = threadIdx.x;                 // 64 threads = 2 waves
  int lane = tid & 31;
  int wave = tid >> 5;
  for (int i = tid; i < DD * DD; i += 64) { wi[i] = W_in[i]; wo[i] = W_out[i]; }
  __syncthreads();

  int row0 = blockIdx.x * ROWS_PER_BLOCK + wave * 16;   // global base row (wave tile)
  float* xw = xs + wave * 16 * DD;
  float* hw = hs + wave * 16 * DD;

  int ln = lane & 15;
  int hi = lane >> 4;                    // 0 or 1

  // constants per lane: condproj fragment (C layout), output bias, x_init
  float cpf[4][8];
  float bo[4];
  float xv[4][8];
#pragma unroll
  for (int n = 0; n < 4; ++n) {
    int col = n * 16 + ln;
    bo[n] = b_out[col];
#pragma unroll
    for (int r = 0; r < 8; ++r) {
      int grow = row0 + r + hi * 8;
      cpf[n][r] = cp[(size_t)grow * DD + col];
      xv[n][r] = rng_normal(1u, (uint32_t)grow, (uint32_t)col);   // x_init ~ N(0,1)
    }
  }

  for (int t = TSTEPS - 1; t >= 0; --t) {
    float sr   = sched[t];
    float srm1 = sched[200 + t];
    float c1   = sched[400 + t];
    float c2   = sched[600 + t];
    float sig  = sched[800 + t];
    float tvv[4];
#pragma unroll
    for (int n = 0; n < 4; ++n) tvv[n] = tv[t * DD + n * 16 + ln];

    // ---- x (C layout regs) -> LDS row-major tile ----
#pragma unroll
    for (int n = 0; n < 4; ++n) {
      int col = n * 16 + ln;
#pragma unroll
      for (int r = 0; r < 8; ++r) xw[(r + hi * 8) * DD + col] = xv[n][r];
    }

    // ---- GEMM1: h_pre = x @ W_in  (16x16 tiles, K chunks of 4) ----
    v2f a[16];
#pragma unroll
    for (int k = 0; k < 16; ++k) {
      int kb = k * 4 + hi * 2;
      a[k] = *(const v2f*)(xw + ln * DD + kb);
    }
    v8f acc[4];
#pragma unroll
    for (int n = 0; n < 4; ++n) acc[n] = vzero8();
#pragma unroll
    for (int k = 0; k < 16; ++k) {
      int kr = k * 4 + hi * 2;
#pragma unroll
      for (int n = 0; n < 4; ++n) {
        int col = n * 16 + ln;
        v2f b;
        b.x = wi[kr * DD + col];
        b.y = wi[(kr + 1) * DD + col];
        acc[n] = __builtin_amdgcn_wmma_f32_16x16x4_f32(
            false, a[k], false, b, (short)0, acc[n], false, false);
      }
    }

    // ---- silu(h_pre + tv[t] + condproj) -> LDS ----
#pragma unroll
    for (int n = 0; n < 4; ++n) {
      int col = n * 16 + ln;
#pragma unroll
      for (int r = 0; r < 8; ++r) {
        float hp = acc[n][r] + tvv[n] + cpf[n][r];
        float hv = hp / (1.0f + __expf(-hp));
        hw[(r + hi * 8) * DD + col] = hv;
      }
    }

    // ---- GEMM2: eps = h @ W_out + b_out ----
#pragma unroll
    for (int k = 0; k < 16; ++k) {
      int kb = k * 4 + hi * 2;
      a[k] = *(const v2f*)(hw + ln * DD + kb);
    }
    v8f acc2[4];
#pragma unroll
    for (int n = 0; n < 4; ++n) acc2[n] = vzero8();
#pragma unroll
    for (int k = 0; k < 16; ++k) {
      int kr = k * 4 + hi * 2;
#pragma unroll
      for (int n = 0; n < 4; ++n) {
        int col = n * 16 + ln;
        v2f b;
        b.x = wo[kr * DD + col];
        b.y = wo[(kr + 1) * DD + col];
        acc2[n] = __builtin_amdgcn_wmma_f32_16x16x4_f32(
            false, a[k], false, b, (short)0, acc2[n], false, false);
      }
    }

    // ---- posterior step: x0 clip, mean, + sigma*noise ----
#pragma unroll
    for (int n = 0; n < 4; ++n) {
      int col = n * 16 + ln;
#pragma unroll
      for (int r = 0; r < 8; ++r) {
        float eps = acc2[n][r] + bo[n];
        float x = xv[n][r];
        float x0 = fminf(1.0f, fmaxf(-1.0f, sr * x - srm1 * eps));
        int grow = row0 + r + hi * 8;
        float nz = rng_normal(2u, (uint32_t)t, (uint32_t)(grow * DD + col));
        xv[n][r] = c1 * x0 + c2 * x + sig * nz;
      }
    }
  }

  // ---- write final sample ----
#pragma unroll
  for (int n = 0; n < 4; ++n) {
    int col = n * 16 + ln;
#pragma unroll
    for (int r = 0; r < 8; ++r) {
      int grow = row0 + r + hi * 8;
      out[(size_t)grow * DD + col] = xv[n][r];
    }
  }
}

extern "C" void kernel_launch(void* const* d_in, const int* in_sizes, int n_in,
                              void* d_out, int out_size, void* d_ws, size_t ws_size,
                              hipStream_t stream) {
  const int*   item_seq = (const int*)d_in[0];
  const float* item_emb = (const float*)d_in[1];
  const float* W_enc1 = (const float*)d_in[2];
  const float* b_enc1 = (const float*)d_in[3];
  const float* W_enc2 = (const float*)d_in[4];
  const float* b_enc2 = (const float*)d_in[5];
  const float* W_in  = (const float*)d_in[6];
  const float* b_in  = (const float*)d_in[7];
  const float* W_t   = (const float*)d_in[8];
  const float* b_t   = (const float*)d_in[9];
  const float* W_c   = (const float*)d_in[10];
  const float* b_c   = (const float*)d_in[11];
  const float* W_out = (const float*)d_in[12];
  const float* b_out = (const float*)d_in[13];
  float* out = (float*)d_out;

  // workspace layout (floats)
  float* ws     = (float*)d_ws;
  float* pooled = ws;                          // 8192*64
  float* cp     = ws + (size_t)BB * DD;        // 8192*64
  float* tv     = cp + (size_t)BB * DD;        // 200*64
  float* sched  = tv + (size_t)TSTEPS * DD;    // 1000

  // K1: gather+pool (one wave per row; 8 waves per 256-thread block)
  gather_pool_kernel<<<BB / 8, 256, 0, stream>>>(item_seq, item_emb, pooled);
  // K2: encoder + condproj
  encoder_kernel<<<BB, 256, 0, stream>>>(pooled, W_enc1, b_enc1, W_enc2, b_enc2,
                                         W_c, b_c, cp);
  // K3: schedule constants
  schedule_kernel<<<1, 32, 0, stream>>>(sched);
  // K4: timestep-embedding projection table
  temb_kernel<<<TSTEPS, DD, 0, stream>>>(W_t, b_t, b_in, tv);
  // K5: persistent WMMA diffusion loop
  diffusion_kernel<<<BB / ROWS_PER_BLOCK, 64, 0, stream>>>(
      W_in, W_out, cp, tv, sched, b_out, out);
}